// KMeansSegmentator_32950989095152
// MI455X (gfx1250) — compile-verified
//
#include <hip/hip_runtime.h>
#include <hip/hip_bf16.h>
#include <stdint.h>

// ---- problem constants (ViT-L/16 kmeans segmentator) ----
#define BS     128
#define NPATCH 196          // 14*14
#define DIM    1024
#define KCENT  64
#define NROWS_ (BS * NPATCH)   // 25088 rows of the GEMM
#define PATCH  16
#define IMG    224
#define PROW   14

#define KC     64           // K-chunk staged in LDS
#define NCHUNK (DIM / KC)   // 16
#define WAVES  8            // waves per block (wave32)
#define ROWS_PER_WAVE 16
#define ROWS_PER_BLK  (WAVES * ROWS_PER_WAVE)   // 128
#define APAD   68           // padded LDS row stride (floats) -> conflict-free b64 frag reads
#define BPAD   68

typedef __attribute__((ext_vector_type(2))) float v2f;
typedef __attribute__((ext_vector_type(8))) float v8f;

// ---- CDNA5 async global->LDS copies (ASYNCcnt path), guarded so the file
// ---- always compiles; fallback is the load+ds_store path.
#if __has_builtin(__builtin_amdgcn_global_load_async_to_lds_b64) && \
    __has_builtin(__builtin_amdgcn_global_load_async_to_lds_b32)
#define USE_ASYNC_LDS 1
#else
#define USE_ASYNC_LDS 0
#endif

// Builtin parameter types (from clang diagnostic): non-const, data-typed,
// gcc-vector pointers in AS1 (global, prints as __device__) / AS3 (LDS).
typedef int i32v2 __attribute__((vector_size(8)));
typedef __attribute__((address_space(1))) i32v2 as1_i32v2;
typedef __attribute__((address_space(3))) i32v2 as3_i32v2;
typedef __attribute__((address_space(1))) int   as1_i32;
typedef __attribute__((address_space(3))) int   as3_i32;

#if USE_ASYNC_LDS
__device__ __forceinline__ void async_g2l_b64(const void* g, void* l) {
    __builtin_amdgcn_global_load_async_to_lds_b64(
        (as1_i32v2*)(uintptr_t)g, (as3_i32v2*)(uint32_t)(uintptr_t)l, 0, 0);
}
__device__ __forceinline__ void async_g2l_b32(const void* g, void* l) {
    __builtin_amdgcn_global_load_async_to_lds_b32(
        (as1_i32*)(uintptr_t)g, (as3_i32*)(uint32_t)(uintptr_t)l, 0, 0);
}
__device__ __forceinline__ void wait_async0() {
#if __has_builtin(__builtin_amdgcn_s_wait_asynccnt)
    __builtin_amdgcn_s_wait_asynccnt(0);
#else
    asm volatile("s_wait_asynccnt 0" ::: "memory");
#endif
}
#endif

// ---------------------------------------------------------------------------
// Kernel 1: xc = feat @ centroids via V_WMMA_F32_16X16X4_F32 (exact f32),
//           fused assignment = argmax_k(0.5*c2[k] - xc[k])  (== argmax l2)
// Double-buffered LDS staging via GLOBAL_LOAD_ASYNC_TO_LDS when available.
// ---------------------------------------------------------------------------
__global__ __launch_bounds__(WAVES * 32)
void kmeans_assign_wmma(const float* __restrict__ feat,
                        const float* __restrict__ centroids,
                        int* __restrict__ assign_out)
{
    __shared__ float ldsA[2][WAVES][ROWS_PER_WAVE][APAD]; // per-wave A tiles (16x64 + pad)
    __shared__ float ldsB[2][KCENT][BPAD];                // B chunk, transposed: [n][k]
    __shared__ float ldsC2[KCENT];                        // 0.5 * sum(centroids[:,n]^2)
    __shared__ float c2part[4][KCENT];                    // c2 partial sums

    const int tid  = threadIdx.x;
    const int wave = tid >> 5;
    const int lane = tid & 31;
    const int half = lane >> 4;        // 0: lanes 0-15, 1: lanes 16-31
    const int l16  = lane & 15;

    const int rowBase = blockIdx.x * ROWS_PER_BLK + wave * ROWS_PER_WAVE;

    // ---- stage one K-chunk into LDS buffer `buf` ----
    auto stage = [&](int kc, int buf) {
        // B chunk transposed: ldsB[buf][n][kk] = centroids[kc+kk][n]
        // 64x64 floats, 256 threads -> 16 each; global reads coalesced per wave.
        #pragma unroll
        for (int i = 0; i < (KC * KCENT) / (WAVES * 32); ++i) {
            int idx = tid + (i << 8);
            int kk  = idx >> 6;
            int n   = idx & 63;
            const float* g = centroids + (size_t)(kc + kk) * KCENT + n;
#if USE_ASYNC_LDS
            async_g2l_b32(g, &ldsB[buf][n][kk]);
#else
            ldsB[buf][n][kk] = *g;
#endif
        }
        // this wave's A tile: 16 rows x 64 k; one 8B element per lane per row.
        #pragma unroll 4
        for (int r = 0; r < ROWS_PER_WAVE; ++r) {
            const float* g = feat + (size_t)(rowBase + r) * DIM + kc + (lane << 1);
#if USE_ASYNC_LDS
            async_g2l_b64(g, &ldsA[buf][wave][r][lane << 1]);
#else
            const float2 v = *(const float2*)g;
            ldsA[buf][wave][r][(lane << 1) + 0] = v.x;
            ldsA[buf][wave][r][(lane << 1) + 1] = v.y;
#endif
        }
    };

    // kick off chunk 0 into buffer 0
    stage(0, 0);

    // ---- 0.5*c2, parallel over all 256 threads (4 segments x 64 cols) ----
    {
        const int col = tid & 63;
        const int seg = tid >> 6;
        float s = 0.f;
        for (int d = seg * (DIM / 4); d < (seg + 1) * (DIM / 4); ++d) {
            float cc = centroids[(size_t)d * KCENT + col];
            s += cc * cc;
        }
        c2part[seg][col] = s;
    }
    __syncthreads();
    if (tid < KCENT)
        ldsC2[tid] = 0.5f * (c2part[0][tid] + c2part[1][tid] +
                             c2part[2][tid] + c2part[3][tid]);

    v8f acc[4] = {};   // 16x64 f32 C tile = 4 x (16x16) tiles

    for (int c = 0; c < NCHUNK; ++c) {
        const int buf = c & 1;
#if USE_ASYNC_LDS
        wait_async0();             // my async copies for chunk c are in LDS
#endif
        __syncthreads();           // WG-wide: buffer `buf` ready, other buffer free

        if (c + 1 < NCHUNK) {
            stage((c + 1) * KC, buf ^ 1);      // overlap next chunk with compute
            if (c + 2 < NCHUNK) {              // pre-pull chunk c+2 toward L2
                const float* nf = feat + (size_t)(rowBase + l16) * DIM
                                  + (c + 2) * KC + (half << 5);
                __builtin_prefetch(nf, 0, 1);  // global_prefetch_b8
            }
        }

        // ---- inner K loop: 16 steps x 4 WMMA (f32, exact) ----
        for (int kk = 0; kk < KC; kk += 4) {
            const int ac = kk + (half << 1);   // lanes 0-15: K=kk,kk+1; 16-31: K=kk+2,kk+3
            // A fragment (16x4): lane l16 holds row M=l16
            v2f a = *(const v2f*)&ldsA[buf][wave][l16][ac];
            #pragma unroll
            for (int t = 0; t < 4; ++t) {
                // B fragment (4x16): VGPR0 = K rows {kk, kk+2}, VGPR1 = {kk+1, kk+3}
                v2f b = *(const v2f*)&ldsB[buf][l16 + (t << 4)][ac];
                acc[t] = __builtin_amdgcn_wmma_f32_16x16x4_f32(
                    /*neg_a=*/false, a, /*neg_b=*/false, b,
                    /*c_mod=*/(short)0, acc[t],
                    /*reuse_a=*/false, /*reuse_b=*/false);
            }
        }
    }

    // ---- epilogue: score = 0.5*c2[n] - xc[n]; argmax over n (first-max wins) ----
    float hc2[4];
    #pragma unroll
    for (int t = 0; t < 4; ++t) hc2[t] = ldsC2[l16 + (t << 4)];

    // C layout: VGPR v, lanes 0-15 -> (M=v, N=lane); lanes 16-31 -> (M=v+8, N=lane-16)
    #pragma unroll
    for (int v = 0; v < 8; ++v) {
        float best  = hc2[0] - acc[0][v];
        int   bestn = l16;
        #pragma unroll
        for (int t = 1; t < 4; ++t) {
            float s = hc2[t] - acc[t][v];
            int   n = l16 + (t << 4);
            if (s > best) { best = s; bestn = n; }
        }
        // reduce across the 16 lanes of each half (xor < 16 stays in-group)
        #pragma unroll
        for (int off = 1; off < 16; off <<= 1) {
            float ob = __shfl_xor(best, off, 32);
            int   on = __shfl_xor(bestn, off, 32);
            if (ob > best || (ob == best && on < bestn)) { best = ob; bestn = on; }
        }
        if (l16 == 0) {
            int row = rowBase + v + (half << 3);
            assign_out[row] = bestn;
        }
    }
}

// ---------------------------------------------------------------------------
// Kernel 2: tile assigned clusters' per-pixel labels into (bs, 224, 224).
// One 224-thread block per output image row -> fully coalesced 896B stores.
// ---------------------------------------------------------------------------
__global__ __launch_bounds__(IMG)
void kmeans_scatter(const float* __restrict__ labels,   // (256, 64)
                    const int* __restrict__ assign,     // (bs*196)
                    float* __restrict__ out)            // (bs, 224, 224)
{
    const int y = blockIdx.x;       // 0..223
    const int b = blockIdx.y;       // 0..127
    const int x = threadIdx.x;      // 0..223

    const int pr = y >> 4, yy = y & 15;
    const int pc = x >> 4, xx = x & 15;

    const int a = assign[b * NPATCH + pr * PROW + pc];
    const float v = labels[(size_t)((yy << 4) + xx) * KCENT + a];
    out[((size_t)b * IMG + y) * IMG + x] = v;
}

// ---------------------------------------------------------------------------
extern "C" void kernel_launch(void* const* d_in, const int* in_sizes, int n_in,
                              void* d_out, int out_size, void* d_ws, size_t ws_size,
                              hipStream_t stream)
{
    const float* feat      = (const float*)d_in[0];   // (128, 196, 1024) f32
    const float* centroids = (const float*)d_in[1];   // (1024, 64) f32
    const float* labels    = (const float*)d_in[2];   // (256, 64) f32
    float* out = (float*)d_out;                       // (128, 224, 224) f32
    int* assign = (int*)d_ws;                         // 25088 ints of scratch

    (void)in_sizes; (void)n_in; (void)out_size; (void)ws_size;

    // GEMM + fused argmax: 196 blocks x 256 threads (8 wave32 waves)
    kmeans_assign_wmma<<<dim3(NROWS_ / ROWS_PER_BLK), dim3(WAVES * 32), 0, stream>>>(
        feat, centroids, assign);

    // Scatter labels into the image grid
    kmeans_scatter<<<dim3(IMG, BS), dim3(IMG), 0, stream>>>(labels, assign, out);
}